// ChamferLoss_11046655885835
// MI455X (gfx1250) — compile-verified
//
#include <hip/hip_runtime.h>

typedef __attribute__((ext_vector_type(2))) float v2f;
typedef __attribute__((ext_vector_type(8))) float v8f;

#define N_PTS 4096
#define WAVES 8     // waves per block (256 threads, wave32)
#define TILE  16

// For each row m of X (per batch), compute min over all rows n of Y of
// ||x_m - y_n||^2 using V_WMMA_F32_16X16X4_F32.
//
// Trick: fold the norms into the K=4 slot so the matrix unit does ALL the
// distance arithmetic:
//   A[m,:] = { x0, x1, x2, 1 }
//   B[:,n] = { -2*y0, -2*y1, -2*y2, ||y||^2 }
//   => D[m][n] = ||y_n||^2 - 2*x_m.y_n
// dist = D + ||x_m||^2, and since ||x_m||^2 is j-invariant it is added once
// AFTER the min reduction. Inner loop = ds_load_b64 + wmma + fmin only.
__global__ __launch_bounds__(256) void chamfer_row_min_kernel(
    const float* __restrict__ X, const float* __restrict__ Y,
    float* __restrict__ outmin)
{
    // All 4096 target points of this batch as float4 {-2y0,-2y1,-2y2,|y|^2}.
    // 64 KB of the 320 KB WGP LDS; 16B stride + 8B half-offset => each
    // wave's b64 load touches all 64 banks exactly once (conflict-free).
    __shared__ float4 ys[N_PTS];

    const int b   = blockIdx.y;
    const int tid = threadIdx.x;

    const float* __restrict__ Yb = Y + (size_t)b * N_PTS * 3;
    #pragma unroll
    for (int k = 0; k < N_PTS / 256; ++k) {
        int n = tid + k * 256;
        float y0 = Yb[n * 3 + 0];
        float y1 = Yb[n * 3 + 1];
        float y2 = Yb[n * 3 + 2];
        float ysq = fmaf(y0, y0, fmaf(y1, y1, y2 * y2));
        ys[n] = make_float4(-2.f * y0, -2.f * y1, -2.f * y2, ysq);
    }
    __syncthreads();

    const int wave = tid >> 5;
    const int lane = tid & 31;
    const int half = lane >> 4;   // 0: K0/K1 half, 1: K2/K3 half
    const int n16  = lane & 15;

    const int i0 = (blockIdx.x * WAVES + wave) * TILE;   // X row-tile base
    const float* __restrict__ Xb = X + (size_t)b * N_PTS * 3;

    // A-matrix (16x4 f32): lanes 0-15 hold {K0,K1} of row M=lane,
    // lanes 16-31 hold {K2,K3} of row M=lane-16. K3 slot carries the 1.
    const int mrow = i0 + n16;
    float x0 = Xb[mrow * 3 + 0];
    float x1 = Xb[mrow * 3 + 1];
    float x2 = Xb[mrow * 3 + 2];
    v2f a;
    a.x = half ? x2  : x0;
    a.y = half ? 1.f : x1;
    float xsq_mine = fmaf(x0, x0, fmaf(x1, x1, x2 * x2));

    // C/D layout: VGPR v holds row M = v + 8*half; grab the 8 row norms of
    // this half (added after the reduction, not inside the loop).
    float xsq[8], rmin[8];
    #pragma unroll
    for (int v = 0; v < 8; ++v) {
        xsq[v]  = __shfl(xsq_mine, 8 * half + v, 32);
        rmin[v] = 3.4e38f;
    }

    // Per-lane B source: lane<16 reads dwords {0,1} of point j0+n16,
    // lane>=16 reads dwords {2,3} — one ds_load_b64, no cndmask.
    const v2f* __restrict__ yss = (const v2f*)ys;
    const int bidx = 2 * n16 + half;

    #pragma unroll 8
    for (int j0 = 0; j0 < N_PTS; j0 += TILE) {
        v2f bb = yss[2 * j0 + bidx];
        v8f c = {};
        c = __builtin_amdgcn_wmma_f32_16x16x4_f32(
            /*neg_a=*/false, a, /*neg_b=*/false, bb,
            /*c_mod=*/(short)0, c, /*reuse_a=*/false, /*reuse_b=*/false);
        #pragma unroll
        for (int v = 0; v < 8; ++v)
            rmin[v] = fminf(rmin[v], c[v]);
    }

    // Min across the 16 lanes of each half (xor masks 1..8 stay in-half).
    #pragma unroll
    for (int v = 0; v < 8; ++v) {
        #pragma unroll
        for (int m = 1; m < 16; m <<= 1)
            rmin[v] = fminf(rmin[v], __shfl_xor(rmin[v], m, 32));
    }

    if (n16 == 0) {   // lanes 0 and 16 each write their half's 8 rows
        float* o = outmin + (size_t)b * N_PTS + i0 + 8 * half;
        #pragma unroll
        for (int v = 0; v < 8; ++v) o[v] = rmin[v] + xsq[v];
    }
}

extern "C" void kernel_launch(void* const* d_in, const int* in_sizes, int n_in,
                              void* d_out, int out_size, void* d_ws, size_t ws_size,
                              hipStream_t stream) {
    const float* points_src = (const float*)d_in[0];
    const float* points_trg = (const float*)d_in[1];
    float* out = (float*)d_out;

    const int B = 16;
    dim3 grid(N_PTS / TILE / WAVES, B);   // 32 x 16 blocks
    dim3 block(256);

    // out[0 : B*N)      = min over dim 1 (over src points) per target point:
    //                     rows = targets, search set = sources.
    chamfer_row_min_kernel<<<grid, block, 0, stream>>>(points_trg, points_src,
                                                       out);
    // out[B*N : 2*B*N)  = min over dim 2 (over trg points) per source point.
    chamfer_row_min_kernel<<<grid, block, 0, stream>>>(points_src, points_trg,
                                                       out + (size_t)B * N_PTS);
}